// SimpPointNet_13305808683485
// MI455X (gfx1250) — compile-verified
//
#include <hip/hip_runtime.h>
#include <hip/hip_bf16.h>
#include <math.h>

typedef __attribute__((ext_vector_type(2))) float v2f;
typedef __attribute__((ext_vector_type(8))) float v8f;

static constexpr int BATCH = 32;
static constexpr int NPTS  = 2048;
static constexpr float BN_EPS   = 1e-5f;
static constexpr float W_THRESH = 1e-4f;

// ---------------------------------------------------------------------------
// Async global -> LDS copy (CDNA5 GLOBAL_LOAD_ASYNC_TO_LDS_B128, ASYNCcnt).
// lds_off = low 32 bits of the flat shared-memory pointer (LDS aperture
// truncation); gaddr = 64-bit global address in a VGPR pair.
// ---------------------------------------------------------------------------
__device__ __forceinline__ void async_copy_b128(void* lds_ptr, const void* gaddr) {
    const unsigned lds_off = (unsigned)(uintptr_t)lds_ptr;
    asm volatile("global_load_async_to_lds_b128 %0, %1, off"
                 :: "v"(lds_off), "v"(gaddr)
                 : "memory");
}

__device__ __forceinline__ void wait_async_all() {
    asm volatile("s_wait_asynccnt 0x0" ::: "memory");
}

// ---------------------------------------------------------------------------
// conv1: 3 -> 64, too small for WMMA; pure VALU, coalesced over n.
// ---------------------------------------------------------------------------
__global__ void conv1_kernel(const float* __restrict__ p,
                             const float* __restrict__ w,
                             const float* __restrict__ bias,
                             float* __restrict__ out) {
    const int n = blockIdx.x * 256 + threadIdx.x;   // grid.x = 8
    const int o = blockIdx.y;                        // 64
    const int b = blockIdx.z;                        // 32
    const size_t pb = (size_t)b * 3 * NPTS;
    float x0 = p[pb + 0 * NPTS + n];
    float x1 = p[pb + 1 * NPTS + n];
    float x2 = p[pb + 2 * NPTS + n];
    out[((size_t)b * 64 + o) * NPTS + n] =
        w[o * 3 + 0] * x0 + w[o * 3 + 1] * x1 + w[o * 3 + 2] * x2 + bias[o];
}

// ---------------------------------------------------------------------------
// WMMA f32 GEMM: Y[b] (O x N) = W (O x C, row stride ldw) * X[b] (C x N) + bias
// Block = 256 threads (8 wave32), 64x64 macro tile, K staged 16-wide into LDS
// with double-buffered GLOBAL_LOAD_ASYNC_TO_LDS_B128 (ASYNCcnt pipelining).
//   sW: [o][k] rows padded to 20 floats -> A fragment is one aligned ds b64,
//       16B-aligned b128 async stores, conflict-free stride-20 reads.
//   sX: [k][n] rows padded to 68 floats -> B fragment via ds_load_2addr pairs.
// ---------------------------------------------------------------------------
__global__ void gemm_wmma_kernel(const float* __restrict__ W, int ldw,
                                 const float* __restrict__ X, int C,
                                 const float* __restrict__ bias, int bias_bstride,
                                 float* __restrict__ Y, int O) {
    __shared__ float sW[2][64][20];   // 10240 B
    __shared__ float sX[2][16][68];   //  8704 B

    const int tid   = threadIdx.x;
    const int lane  = tid & 31;
    const int wave  = tid >> 5;
    const int wm    = wave & 3;          // M sub-tile (16 rows each)
    const int wh    = wave >> 2;         // N half (32 cols each)
    const int l16   = lane & 15;
    const int khalf = (lane >> 4) * 2;   // lanes<16 -> K0/K1, lanes>=16 -> K2/K3

    const int b  = blockIdx.z;
    const int o0 = blockIdx.y * 64;
    const int n0 = blockIdx.x * 64;

    // per-thread staging coordinates
    const int w_kq = tid & 3,  w_o = tid >> 2;   // W tile: 64 rows x 4 k-quads
    const int x_nq = tid & 15, x_c = tid >> 4;   // X tile: 16 ch  x 16 n-quads
    const float* wsrc_base = W + (size_t)(o0 + w_o) * ldw + w_kq * 4;
    const float* xsrc_base = X + ((size_t)b * C + x_c) * NPTS + n0 + x_nq * 4;

    v8f acc0 = {};
    v8f acc1 = {};

    // prologue: stage k0 = 0 into buffer 0
    async_copy_b128(&sW[0][w_o][w_kq * 4], wsrc_base);
    async_copy_b128(&sX[0][x_c][x_nq * 4], xsrc_base);
    wait_async_all();
    __syncthreads();

    int p = 0;
    for (int k0 = 0; k0 < C; k0 += 16) {
        const int k1 = k0 + 16;
        if (k1 < C) {   // uniform predicate: EXEC stays full
            async_copy_b128(&sW[p ^ 1][w_o][w_kq * 4], wsrc_base + k1);
            async_copy_b128(&sX[p ^ 1][x_c][x_nq * 4], xsrc_base + (size_t)k1 * NPTS);
        }

        const float (*sWc)[20] = sW[p];
        const float (*sXc)[68] = sX[p];
#pragma unroll
        for (int kk = 0; kk < 4; ++kk) {
            const int kb = kk * 4 + khalf;
            // A fragment: contiguous b64 from padded [o][k] layout
            v2f a = *(const v2f*)&sWc[wm * 16 + l16][kb];
            // B fragments: 4x16, striped rows
            v2f b0, b1;
            b0.x = sXc[kb + 0][wh * 32 + l16];
            b0.y = sXc[kb + 1][wh * 32 + l16];
            b1.x = sXc[kb + 0][wh * 32 + 16 + l16];
            b1.y = sXc[kb + 1][wh * 32 + 16 + l16];
            acc0 = __builtin_amdgcn_wmma_f32_16x16x4_f32(false, a, false, b0,
                                                         (short)0, acc0, false, false);
            acc1 = __builtin_amdgcn_wmma_f32_16x16x4_f32(false, a, false, b1,
                                                         (short)0, acc1, false, false);
        }

        wait_async_all();    // my prefetches into buffer p^1 have landed
        __syncthreads();     // everyone's landed; everyone done reading p
        p ^= 1;
    }

    // C/D layout: VGPR r -> rows {r, r+8} across lane halves, N = l16
    const int mbase = wm * 16 + (lane >> 4) * 8;
    const int ncol  = n0 + wh * 32 + l16;
#pragma unroll
    for (int r = 0; r < 8; ++r) {
        const int m  = mbase + r;
        const float bv = bias[(size_t)b * bias_bstride + o0 + m];
        const size_t row = ((size_t)b * O + o0 + m) * NPTS;
        Y[row + ncol]      = acc0[r] + bv;
        Y[row + ncol + 16] = acc1[r] + bv;
    }
}

// ---------------------------------------------------------------------------
// Batch-norm stats: one block per channel, reduce over B*N. Produces fused
// scale = g * rsqrt(var+eps), shift = be - scale*mean.
// ---------------------------------------------------------------------------
__global__ void bn_stats_kernel(const float* __restrict__ x, int C,
                                const float* __restrict__ gamma,
                                const float* __restrict__ beta,
                                float* __restrict__ scale,
                                float* __restrict__ shift) {
    const int c   = blockIdx.x;
    const int tid = threadIdx.x;
    float s = 0.f, sq = 0.f;
    for (int i = tid; i < BATCH * NPTS; i += 256) {
        const int bb = i >> 11;
        const int n  = i & (NPTS - 1);
        const float v = x[((size_t)bb * C + c) * NPTS + n];
        s  += v;
        sq += v * v;
    }
    __shared__ float rs[256], rq[256];
    rs[tid] = s; rq[tid] = sq;
    __syncthreads();
    for (int off = 128; off; off >>= 1) {
        if (tid < off) { rs[tid] += rs[tid + off]; rq[tid] += rq[tid + off]; }
        __syncthreads();
    }
    if (tid == 0) {
        const float inv = 1.0f / (float)(BATCH * NPTS);
        const float m   = rs[0] * inv;
        const float var = rq[0] * inv - m * m;
        const float sc  = gamma[c] * rsqrtf(var + BN_EPS);
        scale[c] = sc;
        shift[c] = beta[c] - sc * m;
    }
}

// ---------------------------------------------------------------------------
// BN apply (+ optional ReLU), in place, float4 vectorized.
// ---------------------------------------------------------------------------
template <bool RELU>
__global__ void bn_apply_kernel(float* __restrict__ x,
                                const float* __restrict__ scale,
                                const float* __restrict__ shift, int C) {
    const size_t idx = (size_t)blockIdx.x * 256 + threadIdx.x;  // float4 units
    const int c = (int)((idx / (NPTS / 4)) % C);
    const float sc = scale[c], sh = shift[c];
    float4 v = ((float4*)x)[idx];
    v.x = v.x * sc + sh; v.y = v.y * sc + sh;
    v.z = v.z * sc + sh; v.w = v.w * sc + sh;
    if (RELU) {
        v.x = fmaxf(v.x, 0.f); v.y = fmaxf(v.y, 0.f);
        v.z = fmaxf(v.z, 0.f); v.w = fmaxf(v.w, 0.f);
    }
    ((float4*)x)[idx] = v;
}

// ---------------------------------------------------------------------------
// Channel-wise max over N: one block per (c, b).
// ---------------------------------------------------------------------------
__global__ void max_kernel(const float* __restrict__ x, float* __restrict__ g) {
    const int c = blockIdx.x;   // 1024
    const int b = blockIdx.y;   // 32
    const int tid = threadIdx.x;
    float m = -INFINITY;
    const float* row = x + ((size_t)b * 1024 + c) * NPTS;
    for (int n = tid; n < NPTS; n += 256) m = fmaxf(m, row[n]);
    __shared__ float rm[256];
    rm[tid] = m;
    __syncthreads();
    for (int off = 128; off; off >>= 1) {
        if (tid < off) rm[tid] = fmaxf(rm[tid], rm[tid + off]);
        __syncthreads();
    }
    if (tid == 0) g[(size_t)b * 1024 + c] = rm[0];
}

// ---------------------------------------------------------------------------
// h1 global-feature contribution: h1g[b][o] = h_b1[o] + W1[o, 0:1024] . g[b]
// (folds the broadcast 1024-channel half of the 1088-wide conv into a bias)
// ---------------------------------------------------------------------------
__global__ void h1g_kernel(const float* __restrict__ w1,
                           const float* __restrict__ b1,
                           const float* __restrict__ g,
                           float* __restrict__ out) {
    const int b = blockIdx.y;
    const int o = blockIdx.x * 256 + threadIdx.x;   // grid.x = 2 -> 512
    __shared__ float sg[1024];
    for (int i = threadIdx.x; i < 1024; i += 256) sg[i] = g[(size_t)b * 1024 + i];
    __syncthreads();
    float s = b1[o];
    const float* wr = w1 + (size_t)o * 1088;
    for (int c = 0; c < 1024; ++c) s += wr[c] * sg[c];
    out[(size_t)b * 512 + o] = s;
}

// ---------------------------------------------------------------------------
// Final 128->1 conv + weights = 1 + y. Writes the weights half of d_out.
// ---------------------------------------------------------------------------
__global__ void head_out_kernel(const float* __restrict__ y,
                                const float* __restrict__ w4,
                                const float* __restrict__ b4,
                                float* __restrict__ outw) {
    __shared__ float sw[128];
    if (threadIdx.x < 128) sw[threadIdx.x] = w4[threadIdx.x];
    __syncthreads();
    const int n = blockIdx.x * 256 + threadIdx.x;   // grid.x = 8
    const int b = blockIdx.y;
    float s = b4[0];
    const float* yb = y + (size_t)b * 128 * NPTS + n;
    for (int c = 0; c < 128; ++c) s += sw[c] * yb[(size_t)c * NPTS];
    outw[(size_t)b * NPTS + n] = 1.0f + s;
}

// ---------------------------------------------------------------------------
// Per-batch 3x3 weighted least squares via Cholesky. One block per batch.
// ---------------------------------------------------------------------------
__global__ void solve_kernel(const float* __restrict__ points,
                             const float* __restrict__ wts,
                             float* __restrict__ beta) {
    const int b   = blockIdx.x;
    const int tid = threadIdx.x;
    const float* px = points + (size_t)b * 3 * NPTS;
    const float* wb = wts + (size_t)b * NPTS;

    __shared__ int   ic[256];
    __shared__ float fs[256];
    __shared__ float tot[9];

    // pass 1: valid count
    int cnt = 0;
    for (int n = tid; n < NPTS; n += 256) cnt += (wb[n] > W_THRESH) ? 1 : 0;
    ic[tid] = cnt;
    __syncthreads();
    for (int off = 128; off; off >>= 1) {
        if (tid < off) ic[tid] += ic[tid + off];
        __syncthreads();
    }
    const bool usew = ic[0] > 3;

    // pass 2: accumulate XtWX (sym) and XtWy
    float s0=0,s1=0,s2=0,s3=0,s4=0,s5=0,s6=0,s7=0,s8=0;
    for (int n = tid; n < NPTS; n += 256) {
        const float w  = wb[n];
        const float we = usew ? ((w > W_THRESH) ? w : 0.0f) : 1.0f;
        const float x  = px[n];
        const float y  = px[NPTS + n];
        const float z  = px[2 * NPTS + n];
        s0 += we * x * x;  s1 += we * x * y;  s2 += we * x;
        s3 += we * y * y;  s4 += we * y;      s5 += we;
        s6 += we * x * z;  s7 += we * y * z;  s8 += we * z;
    }
    float loc[9] = {s0,s1,s2,s3,s4,s5,s6,s7,s8};
    for (int j = 0; j < 9; ++j) {
        fs[tid] = loc[j];
        __syncthreads();
        for (int off = 128; off; off >>= 1) {
            if (tid < off) fs[tid] += fs[tid + off];
            __syncthreads();
        }
        if (tid == 0) tot[j] = fs[0];
        __syncthreads();
    }
    if (tid == 0) {
        const float a11 = tot[0], a21 = tot[1], a31 = tot[2];
        const float a22 = tot[3], a32 = tot[4], a33 = tot[5];
        const float b1v = tot[6], b2v = tot[7], b3v = tot[8];
        const float l11 = sqrtf(a11);
        const float l21 = a21 / l11;
        const float l31 = a31 / l11;
        const float l22 = sqrtf(a22 - l21 * l21);
        const float l32 = (a32 - l31 * l21) / l22;
        const float l33 = sqrtf(a33 - l31 * l31 - l32 * l32);
        const float u1 = b1v / l11;
        const float u2 = (b2v - l21 * u1) / l22;
        const float u3 = (b3v - l31 * u1 - l32 * u2) / l33;
        const float be3 = u3 / l33;
        const float be2 = (u2 - l32 * be3) / l22;
        const float be1 = (u1 - l21 * be2 - l31 * be3) / l11;
        beta[(size_t)b * 3 + 0] = be1;
        beta[(size_t)b * 3 + 1] = be2;
        beta[(size_t)b * 3 + 2] = be3;
    }
}

// ---------------------------------------------------------------------------
// Host-side launch orchestration (all on `stream`, graph-capture safe).
// ---------------------------------------------------------------------------
extern "C" void kernel_launch(void* const* d_in, const int* in_sizes, int n_in,
                              void* d_out, int out_size, void* d_ws, size_t ws_size,
                              hipStream_t stream) {
    const float* points = (const float*)d_in[0];
    const float* e_w1 = (const float*)d_in[1];  const float* e_b1 = (const float*)d_in[2];
    const float* e_g1 = (const float*)d_in[3];  const float* e_be1 = (const float*)d_in[4];
    const float* e_w2 = (const float*)d_in[5];  const float* e_b2 = (const float*)d_in[6];
    const float* e_g2 = (const float*)d_in[7];  const float* e_be2 = (const float*)d_in[8];
    const float* e_w3 = (const float*)d_in[9];  const float* e_b3 = (const float*)d_in[10];
    const float* e_g3 = (const float*)d_in[11]; const float* e_be3 = (const float*)d_in[12];
    const float* h_w1 = (const float*)d_in[13]; const float* h_b1 = (const float*)d_in[14];
    const float* h_g1 = (const float*)d_in[15]; const float* h_be1 = (const float*)d_in[16];
    const float* h_w2 = (const float*)d_in[17]; const float* h_b2 = (const float*)d_in[18];
    const float* h_g2 = (const float*)d_in[19]; const float* h_be2 = (const float*)d_in[20];
    const float* h_w3 = (const float*)d_in[21]; const float* h_b3 = (const float*)d_in[22];
    const float* h_g3 = (const float*)d_in[23]; const float* h_be3 = (const float*)d_in[24];
    const float* h_w4 = (const float*)d_in[25]; const float* h_b4 = (const float*)d_in[26];

    float* ws = (float*)d_ws;
    float* out = (float*)d_out;
    float* out_beta = out;          // (32,3)
    float* out_w    = out + 96;     // (32,2048)

    // workspace layout (floats)
    const size_t SZ_X1  = (size_t)BATCH * 64   * NPTS;
    const size_t SZ_T2  = (size_t)BATCH * 128  * NPTS;
    const size_t SZ_BIG = (size_t)BATCH * 1024 * NPTS;
    float* x1  = ws;                                  // pointfeat (lives long)
    float* t2  = x1 + SZ_X1;                          // 128-ch
    float* big = t2 + SZ_T2;                          // 1024-ch, reused below
    float* t4  = big;                                 // 512-ch (after max pool)
    float* t5  = big + (size_t)BATCH * 512 * NPTS;    // 256-ch
    float* t6  = t5  + (size_t)BATCH * 256 * NPTS;    // 128-ch
    float* g   = big + SZ_BIG;                        // (32,1024)
    float* h1g = g + (size_t)BATCH * 1024;            // (32,512)
    float* scale = h1g + (size_t)BATCH * 512;         // (<=1024)
    float* shift = scale + 1024;                      // (<=1024)

    // ---- encoder layer 1: 3 -> 64 ----
    conv1_kernel<<<dim3(NPTS / 256, 64, BATCH), 256, 0, stream>>>(points, e_w1, e_b1, x1);
    bn_stats_kernel<<<dim3(64), 256, 0, stream>>>(x1, 64, e_g1, e_be1, scale, shift);
    bn_apply_kernel<true><<<dim3(BATCH * 64 * (NPTS / 4) / 256), 256, 0, stream>>>(x1, scale, shift, 64);

    // ---- encoder layer 2: 64 -> 128 (WMMA) ----
    gemm_wmma_kernel<<<dim3(NPTS / 64, 128 / 64, BATCH), 256, 0, stream>>>(
        e_w2, 64, x1, 64, e_b2, 0, t2, 128);
    bn_stats_kernel<<<dim3(128), 256, 0, stream>>>(t2, 128, e_g2, e_be2, scale, shift);
    bn_apply_kernel<true><<<dim3(BATCH * 128 * (NPTS / 4) / 256), 256, 0, stream>>>(t2, scale, shift, 128);

    // ---- encoder layer 3: 128 -> 1024 (WMMA), BN no relu ----
    gemm_wmma_kernel<<<dim3(NPTS / 64, 1024 / 64, BATCH), 256, 0, stream>>>(
        e_w3, 128, t2, 128, e_b3, 0, big, 1024);
    bn_stats_kernel<<<dim3(1024), 256, 0, stream>>>(big, 1024, e_g3, e_be3, scale, shift);
    bn_apply_kernel<false><<<dim3(BATCH * 1024 * (NPTS / 4) / 256), 256, 0, stream>>>(big, scale, shift, 1024);

    // ---- global max pool -> g ----
    max_kernel<<<dim3(1024, BATCH), 256, 0, stream>>>(big, g);

    // ---- head layer 1: split 1088 = 1024 (broadcast g, folded to bias) + 64 (pointfeat) ----
    h1g_kernel<<<dim3(2, BATCH), 256, 0, stream>>>(h_w1, h_b1, g, h1g);
    gemm_wmma_kernel<<<dim3(NPTS / 64, 512 / 64, BATCH), 256, 0, stream>>>(
        h_w1 + 1024, 1088, x1, 64, h1g, 512, t4, 512);
    bn_stats_kernel<<<dim3(512), 256, 0, stream>>>(t4, 512, h_g1, h_be1, scale, shift);
    bn_apply_kernel<true><<<dim3(BATCH * 512 * (NPTS / 4) / 256), 256, 0, stream>>>(t4, scale, shift, 512);

    // ---- head layer 2: 512 -> 256 (WMMA) ----
    gemm_wmma_kernel<<<dim3(NPTS / 64, 256 / 64, BATCH), 256, 0, stream>>>(
        h_w2, 512, t4, 512, h_b2, 0, t5, 256);
    bn_stats_kernel<<<dim3(256), 256, 0, stream>>>(t5, 256, h_g2, h_be2, scale, shift);
    bn_apply_kernel<true><<<dim3(BATCH * 256 * (NPTS / 4) / 256), 256, 0, stream>>>(t5, scale, shift, 256);

    // ---- head layer 3: 256 -> 128 (WMMA) ----
    gemm_wmma_kernel<<<dim3(NPTS / 64, 128 / 64, BATCH), 256, 0, stream>>>(
        h_w3, 256, t5, 256, h_b3, 0, t6, 128);
    bn_stats_kernel<<<dim3(128), 256, 0, stream>>>(t6, 128, h_g3, h_be3, scale, shift);
    bn_apply_kernel<true><<<dim3(BATCH * 128 * (NPTS / 4) / 256), 256, 0, stream>>>(t6, scale, shift, 128);

    // ---- head layer 4 + weights output ----
    head_out_kernel<<<dim3(NPTS / 256, BATCH), 256, 0, stream>>>(t6, h_w4, h_b4, out_w);

    // ---- per-batch weighted LS plane fit ----
    solve_kernel<<<dim3(BATCH), 256, 0, stream>>>(points, out_w, out_beta);
}